// graph_encoder_5248450036421
// MI455X (gfx1250) — compile-verified
//
#include <hip/hip_runtime.h>

// MI455X (gfx1250, wave32) implementation.
// Phase 1: zero-fill agg regions of d_out (b128 stores).
// Phase 2: edge scatter via hardware global_atomic_add_f32 (one wave per edge,
//          lane l owns dims 2l..2l+1 -> fully coalesced 256B gathers/scatters;
//          edge metadata read via scalar loads).
// Phase 3: in-place [16,64]x[64,64] tile GEMM with V_WMMA_F32_16X16X4_F32
//          (full fp32 precision; phase is HBM-bound so fp32 WMMA is free),
//          fused sigmoid + mean-of-behaviors epilogue. B fragments hoisted
//          per N-tile so WMMAs issue back-to-back; streaming data uses
//          nontemporal load/store to keep L2 for the embedding tables.

typedef __attribute__((ext_vector_type(2))) float v2f;
typedef __attribute__((ext_vector_type(8))) float v8f;

__device__ __forceinline__ float fast_sigmoid(float x) {
  // v_exp_f32 + v_rcp_f32
  return __builtin_amdgcn_rcpf(1.0f + __expf(-x));
}

__global__ void zero_f4_kernel(float4* __restrict__ p, long long n4) {
  long long i = (long long)blockIdx.x * blockDim.x + threadIdx.x;
  if (i < n4) p[i] = make_float4(0.f, 0.f, 0.f, 0.f);
}

// One wave (32 lanes) per edge. Lane l handles feature dims {2l, 2l+1}.
__global__ void edge_scatter_kernel(const float* __restrict__ user_emb,
                                    const float* __restrict__ item_emb,
                                    const float* __restrict__ vals,  // [B*E]
                                    const int*   __restrict__ rows,  // [B*E]
                                    const int*   __restrict__ cols,  // [B*E]
                                    float* u_agg,   // [B*U,64]
                                    float* i_agg,   // [B*I,64]
                                    int E, int U, int I) {
  int wave = blockIdx.x * (blockDim.x >> 5) + (threadIdx.x >> 5);
  int lane = threadIdx.x & 31;
  if (wave >= 2 * E) return;              // wave-uniform guard
  // wave index is uniform across the 32 lanes: force SGPR so the edge
  // metadata comes in via scalar loads instead of 3 redundant VMEM gathers.
  int wv = __builtin_amdgcn_readfirstlane(wave);
  int b = (wv >= E) ? 1 : 0;              // behavior id (B == 2)
  float v = vals[wv];
  int r = rows[wv];
  int c = cols[wv];
  // coalesced float2 gathers of the 256B embedding rows (L2-resident tables)
  v2f iv = *(const v2f*)(item_emb + (size_t)c * 64 + 2 * lane);
  v2f uv = *(const v2f*)(user_emb + (size_t)r * 64 + 2 * lane);
  float* ud = u_agg + ((size_t)b * U + r) * 64 + 2 * lane;
  float* id = i_agg + ((size_t)b * I + c) * 64 + 2 * lane;
  // hardware f32 atomics (global_atomic_add_f32), resolved at L2
  unsafeAtomicAdd(ud,     v * iv.x);
  unsafeAtomicAdd(ud + 1, v * iv.y);
  unsafeAtomicAdd(id,     v * uv.x);
  unsafeAtomicAdd(id + 1, v * uv.y);
}

// One wave per 16-row tile. Computes z[b] = agg[b, rows] @ W for b in {0,1}
// with V_WMMA_F32_16X16X4_F32 (16 K-steps over K=64, 4 N-tiles), then writes
// sigmoid(z0), sigmoid(z1) in place over agg and sigmoid((z0+z1)/2) to mean.
__global__ void gemm64_sigmoid_kernel(const float* agg,              // [2*U,64] (aliases out_b)
                                      const float* __restrict__ W,   // [64,64] row-major
                                      float* out_b,                  // [2*U,64] in-place
                                      float* __restrict__ out_mean,  // [U,64]
                                      int U) {
  int tile   = blockIdx.x * (blockDim.x >> 5) + (threadIdx.x >> 5);
  int ntiles = U >> 4;
  if (tile >= ntiles) return;             // wave-uniform -> EXEC all ones for WMMA
  int lane = threadIdx.x & 31;
  int half = lane >> 4;                   // K-pair / M-half selector
  int m    = lane & 15;
  size_t row0 = (size_t)tile * 16;

  // Preload full 16x64 A tiles (both behaviors) into VGPRs BEFORE any store:
  // A 16x4 f32 layout: lanes 0-15 hold K={0,1}, lanes 16-31 hold K={2,3}.
  // Nontemporal: each agg element is read exactly once.
  const v2f* base0 = (const v2f*)(agg + (row0 + m) * 64 + 2 * half);
  const v2f* base1 = (const v2f*)(agg + ((size_t)U + row0 + m) * 64 + 2 * half);
  v2f a0[16], a1[16];
#pragma unroll
  for (int kt = 0; kt < 16; ++kt) {
    a0[kt] = __builtin_nontemporal_load(base0 + 2 * kt);
    a1[kt] = __builtin_nontemporal_load(base1 + 2 * kt);
  }

#pragma unroll
  for (int nt = 0; nt < 4; ++nt) {
    // Hoist all 16 B fragments for this N-tile: loads issue as one clause,
    // then the 32 WMMAs chain without per-step loadcnt waits.
    // B 4x16 fragment mirrors A layout: lane half selects K rows {0,1}/{2,3}.
    v2f bfr[16];
#pragma unroll
    for (int kt = 0; kt < 16; ++kt) {
      int krow = 4 * kt + 2 * half;
      bfr[kt].x = W[krow * 64 + nt * 16 + m];
      bfr[kt].y = W[(krow + 1) * 64 + nt * 16 + m];
    }
    v8f c0 = {};
    v8f c1 = {};
#pragma unroll
    for (int kt = 0; kt < 16; ++kt) {
      c0 = __builtin_amdgcn_wmma_f32_16x16x4_f32(false, a0[kt], false, bfr[kt],
                                                 (short)0, c0, false, false);
      c1 = __builtin_amdgcn_wmma_f32_16x16x4_f32(false, a1[kt], false, bfr[kt],
                                                 (short)0, c1, false, false);
    }
    // C/D layout: VGPR v holds M = v + 8*half, N = lane&15.
    // All outputs are write-once streams -> nontemporal stores.
#pragma unroll
    for (int vv = 0; vv < 8; ++vv) {
      size_t rrow = row0 + vv + 8 * half;
      int col = nt * 16 + m;
      float z0 = c0[vv];
      float z1 = c1[vv];
      __builtin_nontemporal_store(fast_sigmoid(z0), &out_b[rrow * 64 + col]);
      __builtin_nontemporal_store(fast_sigmoid(z1),
                                  &out_b[((size_t)U + rrow) * 64 + col]);
      __builtin_nontemporal_store(fast_sigmoid(0.5f * (z0 + z1)),
                                  &out_mean[rrow * 64 + col]);
    }
  }
}

extern "C" void kernel_launch(void* const* d_in, const int* in_sizes, int n_in,
                              void* d_out, int out_size, void* d_ws, size_t ws_size,
                              hipStream_t stream) {
  const float* user_emb = (const float*)d_in[0];  // [U,64]
  const float* item_emb = (const float*)d_in[1];  // [I,64]
  const float* u_w      = (const float*)d_in[2];  // [64,64]
  const float* i_w      = (const float*)d_in[3];  // [64,64]
  const float* vals     = (const float*)d_in[4];  // [B,E]
  const int*   rows     = (const int*)d_in[5];    // [B,E]
  const int*   cols     = (const int*)d_in[6];    // [B,E]
  float* out = (float*)d_out;

  const int D = 64, B = 2;
  const int U = in_sizes[0] / D;
  const int I = in_sizes[1] / D;
  const int E = in_sizes[4] / B;

  // d_out layout (reference return order):
  float* user_embedding  = out;                                   // [U,64]
  float* item_embedding  = user_embedding + (size_t)U * D;        // [I,64]
  float* user_embeddings = item_embedding + (size_t)I * D;        // [B,U,64] (= u_agg)
  float* item_embeddings = user_embeddings + (size_t)B * U * D;   // [B,I,64] (= i_agg)

  // Phase 1: zero the accumulation regions (contiguous span).
  long long n4 = ((long long)B * U * D + (long long)B * I * D) / 4;
  zero_f4_kernel<<<(unsigned)((n4 + 255) / 256), 256, 0, stream>>>(
      (float4*)user_embeddings, n4);

  // Phase 2: edge scatter (8 waves = 8 edges per 256-thread block).
  int total_waves = B * E;
  edge_scatter_kernel<<<(total_waves + 7) / 8, 256, 0, stream>>>(
      user_emb, item_emb, vals, rows, cols,
      user_embeddings, item_embeddings, E, U, I);

  // Phase 3: WMMA GEMM + sigmoid (+mean), in place (8 tiles per block).
  int tilesU = U / 16;
  gemm64_sigmoid_kernel<<<(tilesU + 7) / 8, 256, 0, stream>>>(
      user_embeddings, u_w, user_embeddings, user_embedding, U);
  int tilesI = I / 16;
  gemm64_sigmoid_kernel<<<(tilesI + 7) / 8, 256, 0, stream>>>(
      item_embeddings, i_w, item_embeddings, item_embedding, I);
}